// GCN_79499844649691
// MI455X (gfx1250) — compile-verified
//
#include <hip/hip_runtime.h>

typedef __attribute__((ext_vector_type(2))) float v2f;
typedef __attribute__((ext_vector_type(8))) float v8f;

#define EPSV 1e-5f

// ---------------------------------------------------------------------------
// GEMM: H[M,Cout] = X[M,K] * W[K,Cout], fp32 WMMA 16x16x4, one wave per tile.
// ISA layouts (CDNA5 ISA §7.12.2):
//   A (16x4, MxK): VGPR r, lanes 0-15: K=r, M=lane; lanes 16-31: K=r+2.
//   B (4x16, KxN): VGPR r, lanes 0-15: K=r, N=lane; lanes 16-31: K=r+2.
//   D (16x16):     VGPR r, lanes 0-15: M=r, N=lane; lanes 16-31: M=r+8.
// ---------------------------------------------------------------------------
__global__ void gemm_wmma_f32(const float* __restrict__ X,
                              const float* __restrict__ W,
                              float* __restrict__ H,
                              int M, int K, int Cout) {
  const int nTiles = Cout >> 4;
  const int totalTiles = (M >> 4) * nTiles;
  const int wave = blockIdx.x * (blockDim.x >> 5) + (threadIdx.x >> 5);
  if (wave >= totalTiles) return;                 // wave-uniform: EXEC stays all-1s
  const int lane = threadIdx.x & 31;
  const int ln = lane & 15;
  const int g  = lane >> 4;                      // lane half
  const int g2 = g << 1;                         // K offset 0 or 2
  const int mTile = wave / nTiles;
  const int nTile = wave - mTile * nTiles;
  const int m0 = mTile << 4, n0 = nTile << 4;

  const float* __restrict__ xrow = X + (m0 + ln) * K;   // A row m0+ln
  const float* __restrict__ wcol = W + (n0 + ln);       // B col n0+ln

  v8f acc = {};
  for (int k = 0; k < K; k += 4) {
    v2f a, b;
    a.x = xrow[k + g2];
    a.y = xrow[k + g2 + 1];
    b.x = wcol[(k + g2) * Cout];
    b.y = wcol[(k + g2 + 1) * Cout];
    acc = __builtin_amdgcn_wmma_f32_16x16x4_f32(
        /*neg_a=*/false, a, /*neg_b=*/false, b,
        /*c_mod=*/(short)0, acc, /*reuse_a=*/false, /*reuse_b=*/false);
  }

  const int mg = m0 + (g << 3);
#pragma unroll
  for (int r = 0; r < 8; ++r)
    H[(mg + r) * Cout + n0 + ln] = acc[r];
}

// ---------------------------------------------------------------------------
// Degree / normalization helpers
// ---------------------------------------------------------------------------
__global__ void deg_init(float* deg, int n) {
  int t = blockIdx.x * blockDim.x + threadIdx.x;
  if (t < n) deg[t] = 1.0f;                     // self-loop contributes 1
}
__global__ void deg_accum(const int* __restrict__ dst, float* deg, int E) {
  int e = blockIdx.x * blockDim.x + threadIdx.x;
  if (e < E)
    __hip_atomic_fetch_add(&deg[dst[e]], 1.0f, __ATOMIC_RELAXED,
                           __HIP_MEMORY_SCOPE_AGENT);
}
__global__ void deg_rsqrt(float* deg, int n) {
  int t = blockIdx.x * blockDim.x + threadIdx.x;
  if (t < n) deg[t] = rsqrtf(deg[t]);
}

// ---------------------------------------------------------------------------
// agg[i,c] = h[i,c]*dinv[i]^2 (self-loop message) + bias[c]
// ---------------------------------------------------------------------------
template <int C>
__global__ void agg_init(const float* __restrict__ h,
                         const float* __restrict__ dinv,
                         const float* __restrict__ bias,
                         float* __restrict__ agg, int n) {
  constexpr int L = (C == 64) ? 6 : 7;
  int t = blockIdx.x * blockDim.x + threadIdx.x;
  if (t >= (n << L)) return;
  int i = t >> L;
  int c = t & (C - 1);
  float di = dinv[i];
  agg[t] = h[t] * di * di + bias[c];
}

// ---------------------------------------------------------------------------
// Edge scatter: agg[dst] += h[src] * dinv[src]*dinv[dst]
// float4 row-segment loads (global_load_b128) + fp32 L2 atomics.
// ---------------------------------------------------------------------------
template <int C>
__global__ void edge_scatter(const float* __restrict__ h,
                             const int* __restrict__ src,
                             const int* __restrict__ dst,
                             const float* __restrict__ dinv,
                             float* __restrict__ agg, int E) {
  constexpr int Q = C / 4;
  int t = blockIdx.x * blockDim.x + threadIdx.x;
  if (t >= E * Q) return;
  int e = t / Q;
  int q = t - e * Q;
  int s = src[e], d = dst[e];
  float nrm = dinv[s] * dinv[d];
  float4 v = reinterpret_cast<const float4*>(h)[s * Q + q];
  float* a = agg + d * C + (q << 2);
  __hip_atomic_fetch_add(a + 0, v.x * nrm, __ATOMIC_RELAXED, __HIP_MEMORY_SCOPE_AGENT);
  __hip_atomic_fetch_add(a + 1, v.y * nrm, __ATOMIC_RELAXED, __HIP_MEMORY_SCOPE_AGENT);
  __hip_atomic_fetch_add(a + 2, v.z * nrm, __ATOMIC_RELAXED, __HIP_MEMORY_SCOPE_AGENT);
  __hip_atomic_fetch_add(a + 3, v.w * nrm, __ATOMIC_RELAXED, __HIP_MEMORY_SCOPE_AGENT);
}

// ---------------------------------------------------------------------------
// BatchNorm: per-channel sum / sumsq with LDS reduction, then apply.
// ---------------------------------------------------------------------------
__global__ void zero_stats(float* stats) {        // 256 floats: sum[128]+sumsq[128]
  stats[threadIdx.x] = 0.0f;
}

template <int C>
__global__ void bn_stats(const float* __restrict__ x, float* __restrict__ sums,
                         float* __restrict__ sumsq, int n) {
  constexpr int R = 256 / C;                      // node rows per block pass
  __shared__ float sm[256], sq[256];
  int c = threadIdx.x & (C - 1);
  int r = threadIdx.x / C;
  float s = 0.f, s2 = 0.f;
  for (int i = blockIdx.x * R + r; i < n; i += gridDim.x * R) {
    float v = x[i * C + c];
    s += v;
    s2 += v * v;
  }
  sm[threadIdx.x] = s;
  sq[threadIdx.x] = s2;
  __syncthreads();
  if (threadIdx.x < C) {
    float ts = 0.f, t2 = 0.f;
#pragma unroll
    for (int j = 0; j < R; ++j) {
      ts += sm[j * C + c];
      t2 += sq[j * C + c];
    }
    __hip_atomic_fetch_add(&sums[c], ts, __ATOMIC_RELAXED, __HIP_MEMORY_SCOPE_AGENT);
    __hip_atomic_fetch_add(&sumsq[c], t2, __ATOMIC_RELAXED, __HIP_MEMORY_SCOPE_AGENT);
  }
}

template <int C, bool RELU>
__global__ void bn_apply(const float* __restrict__ x,
                         const float* __restrict__ sums,
                         const float* __restrict__ sumsq,
                         const float* __restrict__ gamma,
                         const float* __restrict__ beta,
                         float* __restrict__ y, int n) {
  int t = blockIdx.x * blockDim.x + threadIdx.x;
  if (t >= n * C) return;
  int c = t & (C - 1);
  float invn = 1.0f / (float)n;
  float m = sums[c] * invn;
  float var = fmaxf(sumsq[c] * invn - m * m, 0.0f);
  float sc = rsqrtf(var + EPSV) * gamma[c];
  float v = (x[t] - m) * sc + beta[c];
  if (RELU) v = fmaxf(v, 0.0f);
  y[t] = v;
}

// ---------------------------------------------------------------------------
// Host orchestration
// ---------------------------------------------------------------------------
static inline int cdiv(int a, int b) { return (a + b - 1) / b; }

extern "C" void kernel_launch(void* const* d_in, const int* in_sizes, int n_in,
                              void* d_out, int out_size, void* d_ws, size_t ws_size,
                              hipStream_t stream) {
  const float* x  = (const float*)d_in[0];
  const int*   ei = (const int*)d_in[1];
  const float* W1 = (const float*)d_in[2];  const float* b1 = (const float*)d_in[3];
  const float* g1 = (const float*)d_in[4];  const float* be1 = (const float*)d_in[5];
  const float* W2 = (const float*)d_in[6];  const float* b2 = (const float*)d_in[7];
  const float* g2 = (const float*)d_in[8];  const float* be2 = (const float*)d_in[9];
  const float* W3 = (const float*)d_in[10]; const float* b3 = (const float*)d_in[11];
  const float* g3 = (const float*)d_in[12]; const float* be3 = (const float*)d_in[13];

  const int n = in_sizes[0] / 64;        // 50000
  const int E = in_sizes[1] / 2;         // 800000
  const int* src = ei;
  const int* dst = ei + E;

  // workspace carve-out
  char* ws = (char*)d_ws;
  size_t off = ((size_t)n * 4 + 511) & ~(size_t)511;
  float* dinv  = (float*)ws;
  float* bufA  = (float*)(ws + off);               // n*128 floats
  float* bufB  = bufA + (size_t)n * 128;           // n*128 floats
  float* sums  = bufB + (size_t)n * 128;           // 128
  float* sumsq = sums + 128;                       // 128 (contiguous with sums)
  float* outp  = (float*)d_out;

  const int TB = 256;

  // degrees -> dinv
  deg_init<<<cdiv(n, TB), TB, 0, stream>>>(dinv, n);
  deg_accum<<<cdiv(E, TB), TB, 0, stream>>>(dst, dinv, E);
  deg_rsqrt<<<cdiv(n, TB), TB, 0, stream>>>(dinv, n);

  auto gemm = [&](const float* A, const float* W, float* H, int K, int Cout) {
    int tiles = (n / 16) * (Cout / 16);
    gemm_wmma_f32<<<cdiv(tiles, 4), 128, 0, stream>>>(A, W, H, n, K, Cout);
  };

  // ---- Layer 1: x[·,64] -> bufA; agg -> bufB; bn+relu -> bufA
  gemm(x, W1, bufA, 64, 64);
  agg_init<64><<<cdiv(n * 64, TB), TB, 0, stream>>>(bufA, dinv, b1, bufB, n);
  edge_scatter<64><<<cdiv(E * 16, TB), TB, 0, stream>>>(bufA, src, dst, dinv, bufB, E);
  zero_stats<<<1, 256, 0, stream>>>(sums);
  bn_stats<64><<<1024, 256, 0, stream>>>(bufB, sums, sumsq, n);
  bn_apply<64, true><<<cdiv(n * 64, TB), TB, 0, stream>>>(bufB, sums, sumsq, g1, be1, bufA, n);

  // ---- Layer 2: bufA[·,64] -> bufB[·,128]; agg -> bufA; bn+relu -> bufB
  gemm(bufA, W2, bufB, 64, 128);
  agg_init<128><<<cdiv(n * 128, TB), TB, 0, stream>>>(bufB, dinv, b2, bufA, n);
  edge_scatter<128><<<cdiv(E * 32, TB), TB, 0, stream>>>(bufB, src, dst, dinv, bufA, E);
  zero_stats<<<1, 256, 0, stream>>>(sums);
  bn_stats<128><<<1024, 256, 0, stream>>>(bufA, sums, sumsq, n);
  bn_apply<128, true><<<cdiv(n * 128, TB), TB, 0, stream>>>(bufA, sums, sumsq, g2, be2, bufB, n);

  // ---- Layer 3: bufB[·,128] -> bufA[·,64]; agg -> bufB; bn -> d_out
  gemm(bufB, W3, bufA, 128, 64);
  agg_init<64><<<cdiv(n * 64, TB), TB, 0, stream>>>(bufA, dinv, b3, bufB, n);
  edge_scatter<64><<<cdiv(E * 16, TB), TB, 0, stream>>>(bufA, src, dst, dinv, bufB, E);
  zero_stats<<<1, 256, 0, stream>>>(sums);
  bn_stats<64><<<1024, 256, 0, stream>>>(bufB, sums, sumsq, n);
  bn_apply<64, false><<<cdiv(n * 64, TB), TB, 0, stream>>>(bufB, sums, sumsq, g3, be3, outp, n);

  (void)n_in; (void)out_size; (void)ws_size;
}